// DegreeDeviation_87832081203329
// MI455X (gfx1250) — compile-verified
//
#include <hip/hip_runtime.h>
#include <hip/hip_bf16.h>
#include <math.h>

typedef __attribute__((ext_vector_type(2))) float v2f;
typedef __attribute__((ext_vector_type(8))) float v8f;

#define BLOCK 256
#define RED_BLOCKS 256
#define WAVES_PER_BLOCK (BLOCK / 32)

// ---------------------------------------------------------------------------
// Kernel 1: zero the degree counters (stored as u32 in d_out). 128-bit stores.
// ---------------------------------------------------------------------------
__global__ void dd_zero_kernel(unsigned* counts, int n4, int n) {
    int i = blockIdx.x * blockDim.x + threadIdx.x;
    if (i < n4) {
        uint4 z = {0u, 0u, 0u, 0u};
        reinterpret_cast<uint4*>(counts)[i] = z;
    }
    int rem = n - (n4 << 2);
    if (i < rem) counts[(n4 << 2) + i] = 0u;
}

// ---------------------------------------------------------------------------
// Kernel 2: scatter-add. 128-bit coalesced loads of 4 edge endpoints per
// thread, then 4 no-return u32 atomics (global_atomic_add_u32, STOREcnt path).
// Counters (4 MB) are L2-resident; random indices => negligible contention.
// This kernel carries the roofline: 256 MB edge reads ~ 11 us @ 23.3 TB/s.
// ---------------------------------------------------------------------------
__global__ void dd_scatter_kernel(const int* __restrict__ edges,
                                  unsigned* __restrict__ counts,
                                  int n4, int total) {
    int i = blockIdx.x * blockDim.x + threadIdx.x;
    if (i < n4) {
        int4 e = reinterpret_cast<const int4*>(edges)[i];
        atomicAdd(&counts[e.x], 1u);
        atomicAdd(&counts[e.y], 1u);
        atomicAdd(&counts[e.z], 1u);
        atomicAdd(&counts[e.w], 1u);
    }
    int rem = total - (n4 << 2);
    if (i < rem) {
        atomicAdd(&counts[edges[(n4 << 2) + i]], 1u);
    }
}

// ---------------------------------------------------------------------------
// Kernel 3: per-block partial (sum, sumsq) via v_wmma_f32_16x16x4_f32.
// B == ones => sum(D) = 16 * sum(A). Main loop covers only FULL 64-element
// tiles: unguarded uint2 load (one global_load_b64/lane), wave-uniform trip
// count, EXEC stays all-ones for WMMA. The <64-element boundary tail is
// added with scalar FLOPs by global wave 0 (scaled correctly: WMMA part /16
// before combining). Fixed shapes + fixed order => deterministic.
// ---------------------------------------------------------------------------
__global__ void dd_partial_kernel(const unsigned* __restrict__ counts,
                                  float* __restrict__ partials, int n) {
    const int tid    = threadIdx.x;
    const int lane   = tid & 31;
    const int wave   = tid >> 5;
    const int gwave  = blockIdx.x * WAVES_PER_BLOCK + wave;
    const int nwaves = gridDim.x * WAVES_PER_BLOCK;
    const int full   = n >> 6;          // # of complete 64-element tiles

    v8f accS = {};
    v8f accQ = {};
    v2f ones;
    ones[0] = 1.0f;
    ones[1] = 1.0f;

    const uint2* c2 = reinterpret_cast<const uint2*>(counts);
    for (int t = gwave; t < full; t += nwaves) {     // wave-uniform
        uint2 c  = c2[(t << 5) + lane];              // global_load_b64
        float x0 = (float)c.x;
        float x1 = (float)c.y;
        v2f a;  a[0]  = x0;      a[1]  = x1;
        v2f a2; a2[0] = x0 * x0; a2[1] = x1 * x1;
        accS = __builtin_amdgcn_wmma_f32_16x16x4_f32(
            false, a,  false, ones, (short)0, accS, false, false);
        accQ = __builtin_amdgcn_wmma_f32_16x16x4_f32(
            false, a2, false, ones, (short)0, accQ, false, false);
    }

    float s = 0.0f, q = 0.0f;
#pragma unroll
    for (int i = 0; i < 8; ++i) { s += accS[i]; q += accQ[i]; }
    // D replicates row sums across 16 columns -> scale WMMA part by 1/16
    s *= 0.0625f;
    q *= 0.0625f;

    // boundary tail (< 64 elements), handled once globally by wave 0
    if (gwave == 0) {
        int idx = (full << 6) + (lane << 1);
        if (idx < n) {
            float x = (float)counts[idx];
            s += x;
            q += x * x;
        }
        if (idx + 1 < n) {
            float x = (float)counts[idx + 1];
            s += x;
            q += x * x;
        }
    }

#pragma unroll
    for (int d = 16; d >= 1; d >>= 1) {
        s += __shfl_xor(s, d, 32);
        q += __shfl_xor(q, d, 32);
    }

    __shared__ float ls[WAVES_PER_BLOCK], lq[WAVES_PER_BLOCK];
    if (lane == 0) { ls[wave] = s; lq[wave] = q; }
    __syncthreads();
    if (tid == 0) {
        float S = 0.0f, Q = 0.0f;
        for (int w = 0; w < WAVES_PER_BLOCK; ++w) { S += ls[w]; Q += lq[w]; }
        partials[2 * blockIdx.x]     = S;
        partials[2 * blockIdx.x + 1] = Q;
    }
}

// ---------------------------------------------------------------------------
// Kernel 4: single-block double-precision tree reduction over the partials;
// writes mean and 1/(std+1e-8) (std unbiased, ddof=1) to scalars.
// ---------------------------------------------------------------------------
__global__ void dd_finalize_kernel(const float* __restrict__ partials, int nb,
                                   float* __restrict__ scalars, int n) {
    __shared__ double sh_s[256], sh_q[256];
    int tid = threadIdx.x;
    double s = 0.0, q = 0.0;
    for (int i = tid; i < nb; i += 256) {
        s += (double)partials[2 * i];
        q += (double)partials[2 * i + 1];
    }
    sh_s[tid] = s;
    sh_q[tid] = q;
    __syncthreads();
    for (int d = 128; d >= 1; d >>= 1) {
        if (tid < d) {
            sh_s[tid] += sh_s[tid + d];
            sh_q[tid] += sh_q[tid + d];
        }
        __syncthreads();
    }
    if (tid == 0) {
        double N    = (double)n;
        double sum  = sh_s[0];
        double sq   = sh_q[0];
        double mean = sum / N;
        double var  = (sq - sum * sum / N) / (N - 1.0);
        if (var < 0.0) var = 0.0;
        double sd   = sqrt(var) + 1e-8;
        scalars[0]  = (float)mean;
        scalars[1]  = (float)(1.0 / sd);
    }
}

// ---------------------------------------------------------------------------
// Kernel 5: in-place normalize: 128-bit read of u32 counts, 128-bit write of
// f32 results to the same buffer (each thread owns its 16B slot => safe).
// ---------------------------------------------------------------------------
__global__ void dd_normalize_kernel(const unsigned* counts, float* out,
                                    const float* __restrict__ scalars,
                                    int n4, int n) {
    int i = blockIdx.x * blockDim.x + threadIdx.x;
    float mean = scalars[0];
    float istd = scalars[1];
    if (i < n4) {
        uint4 c = reinterpret_cast<const uint4*>(counts)[i];
        float4 r;
        r.x = ((float)c.x - mean) * istd;
        r.y = ((float)c.y - mean) * istd;
        r.z = ((float)c.z - mean) * istd;
        r.w = ((float)c.w - mean) * istd;
        reinterpret_cast<float4*>(out)[i] = r;
    }
    int rem = n - (n4 << 2);
    if (i < rem) {
        int idx  = (n4 << 2) + i;
        out[idx] = ((float)counts[idx] - mean) * istd;
    }
}

// ---------------------------------------------------------------------------
extern "C" void kernel_launch(void* const* d_in, const int* in_sizes, int n_in,
                              void* d_out, int out_size, void* d_ws, size_t ws_size,
                              hipStream_t stream) {
    const int* edges = (const int*)d_in[0];   // int32 edge endpoints, flat
    const int  total = in_sizes[0];           // 2 * NUM_EDGES node indices
    const int  n     = out_size;              // NUM_NODES

    unsigned* counts   = (unsigned*)d_out;    // counters live in d_out
    float*    outf     = (float*)d_out;
    float*    partials = (float*)d_ws;                 // 2*RED_BLOCKS floats
    float*    scalars  = partials + 2 * RED_BLOCKS;    // mean, istd

    const int nn4   = n >> 2;
    const int nrem  = n - (nn4 << 2);
    const int zwork = (nn4 > nrem) ? nn4 : nrem;
    const int zb    = (zwork + BLOCK - 1) / BLOCK;
    dd_zero_kernel<<<zb, BLOCK, 0, stream>>>(counts, nn4, n);

    const int e4    = total >> 2;
    const int erem  = total - (e4 << 2);
    const int swork = (e4 > erem) ? e4 : erem;
    const int sb    = (swork + BLOCK - 1) / BLOCK;
    dd_scatter_kernel<<<sb, BLOCK, 0, stream>>>(edges, counts, e4, total);

    dd_partial_kernel<<<RED_BLOCKS, BLOCK, 0, stream>>>(counts, partials, n);
    dd_finalize_kernel<<<1, 256, 0, stream>>>(partials, RED_BLOCKS, scalars, n);
    dd_normalize_kernel<<<zb, BLOCK, 0, stream>>>(counts, outf, scalars, nn4, n);
}